// SSAccuracyLoss3_46325517254720
// MI455X (gfx1250) — compile-verified
//
#include <hip/hip_runtime.h>

typedef __attribute__((ext_vector_type(4))) float v4f;
typedef __attribute__((ext_vector_type(2))) float v2f;
typedef __attribute__((ext_vector_type(8))) float v8f;

#define N_ITEMS (256 * 8192)   // B*L rows, each row has C=8 contiguous floats

__global__ void ssacc_zero_ws(float* ws) {
    ws[0] = 0.0f;
    ws[1] = 0.0f;
}

__global__ __launch_bounds__(256) void ssacc_main(const float* __restrict__ real,
                                                  const float* __restrict__ pred,
                                                  float* __restrict__ ws) {
    const v4f* r4 = (const v4f*)real;
    const v4f* p4 = (const v4f*)pred;
    const int tid    = blockIdx.x * blockDim.x + threadIdx.x;
    const int stride = gridDim.x * blockDim.x;

    float c_arg = 0.0f;   // count of argmax(pred)==argmax(real)
    float c_lbl = 0.0f;   // count of exact threshold match

    for (int t = tid; t < N_ITEMS; t += stride) {
        // one (b,l) row = 8 floats = 2x b128 loads per tensor; lanes contiguous
        // -> fully coalesced. Single-use stream -> non-temporal.
        v4f p0 = __builtin_nontemporal_load(p4 + 2 * t);
        v4f p1 = __builtin_nontemporal_load(p4 + 2 * t + 1);
        v4f r0 = __builtin_nontemporal_load(r4 + 2 * t);
        v4f r1 = __builtin_nontemporal_load(r4 + 2 * t + 1);

        float p[8] = {p0.x, p0.y, p0.z, p0.w, p1.x, p1.y, p1.z, p1.w};
        float r[8] = {r0.x, r0.y, r0.z, r0.w, r1.x, r1.y, r1.z, r1.w};

        // first-index argmax semantics (strict >) for both tensors
        int   bi = 0, ti = 0;
        float bp = p[0], br = r[0];
        bool  ok = true;
#pragma unroll
        for (int c = 1; c < 8; ++c) {
            if (p[c] > bp) { bp = p[c]; bi = c; }
            if (r[c] > br) { br = r[c]; ti = c; }
        }
#pragma unroll
        for (int c = 0; c < 8; ++c) {
            ok = ok && (r[c] == ((p[c] > 0.5f) ? 1.0f : 0.0f));
        }
        c_arg += (bi == ti) ? 1.0f : 0.0f;
        c_lbl += ok ? 1.0f : 0.0f;
    }

    // ---- wave32 reduction via V_WMMA_F32_16X16X4_F32 ----
    // A layout (16x4 f32): VGPR0 = {K0: lanes0-15, K2: lanes16-31},
    //                      VGPR1 = {K1, K3}. Put partial in K0/K2 slot, zero K1/K3.
    // B = all ones (layout-independent), C = 0:
    //   D[m,n] = c(lane m) + c(lane m+16)   for every n.
    // Summing the 8 D VGPRs gives each half-wave the sum of 16 lanes;
    // shfl_xor(16) completes the 32-lane sum.
    v2f bones; bones.x = 1.0f; bones.y = 1.0f;
    v8f czero = {};

    v2f a0; a0.x = c_arg; a0.y = 0.0f;
    v8f d0 = __builtin_amdgcn_wmma_f32_16x16x4_f32(false, a0, false, bones,
                                                   (short)0, czero, false, false);
    v2f a1; a1.x = c_lbl; a1.y = 0.0f;
    v8f d1 = __builtin_amdgcn_wmma_f32_16x16x4_f32(false, a1, false, bones,
                                                   (short)0, czero, false, false);

    float s0 = d0[0] + d0[1] + d0[2] + d0[3] + d0[4] + d0[5] + d0[6] + d0[7];
    float s1 = d1[0] + d1[1] + d1[2] + d1[3] + d1[4] + d1[5] + d1[6] + d1[7];
    s0 += __shfl_xor(s0, 16, 32);
    s1 += __shfl_xor(s1, 16, 32);

    // ---- block reduction (8 waves) + one pair of global atomics per block ----
    __shared__ float red[16];
    const int wave = threadIdx.x >> 5;
    const int lane = threadIdx.x & 31;
    if (lane == 0) {
        red[wave * 2 + 0] = s0;
        red[wave * 2 + 1] = s1;
    }
    __syncthreads();
    if (threadIdx.x == 0) {
        float t0 = 0.0f, t1 = 0.0f;
#pragma unroll
        for (int w = 0; w < 8; ++w) {
            t0 += red[w * 2 + 0];
            t1 += red[w * 2 + 1];
        }
        // counts are exact integers < 2^24 -> float atomic adds are exact
        // and order-independent -> deterministic output.
        atomicAdd(&ws[0], t0);
        atomicAdd(&ws[1], t1);
    }
}

__global__ void ssacc_finalize(const float* __restrict__ ws, float* __restrict__ out) {
    const float invN = 1.0f / (float)N_ITEMS;
    const float acc_max = ws[0] * invN;
    const float acc_lbl = ws[1] * invN;
    out[0] = 0.5f * (1.0f - acc_max) + 0.5f * (1.0f - acc_lbl);
}

extern "C" void kernel_launch(void* const* d_in, const int* in_sizes, int n_in,
                              void* d_out, int out_size, void* d_ws, size_t ws_size,
                              hipStream_t stream) {
    const float* real = (const float*)d_in[0];  // [B, L, C] one-hot f32
    const float* pred = (const float*)d_in[1];  // [B, L, C] f32
    float* ws  = (float*)d_ws;
    float* out = (float*)d_out;

    ssacc_zero_ws<<<1, 1, 0, stream>>>(ws);

    // 1024 blocks * 256 threads = 262144 threads, 8 rows (512 B) per thread.
    // 8192 waves in flight -> enough MLP to saturate 23.3 TB/s.
    ssacc_main<<<1024, 256, 0, stream>>>(real, pred, ws);

    ssacc_finalize<<<1, 1, 0, stream>>>(ws, out);
}